// RNAPocketEncoderV3_3547642987459
// MI455X (gfx1250) — compile-verified
//
#include <hip/hip_runtime.h>
#include <stdint.h>

// ---- problem constants (from reference) ----
#define N_SCALAR 32
#define N_VEC    16
#define VEC_DIM  3
#define N_TEN    8
#define TEN_DIM  5
#define FEAT     (N_SCALAR + N_VEC * VEC_DIM + N_TEN * TEN_DIM)   // 120
#define EPS      1e-5f

// ---- tiling ----
#define TILE 128          // rows per workgroup == blockDim.x (1 row / thread)

typedef unsigned int u32x4 __attribute__((ext_vector_type(4)));
typedef int          i32x8 __attribute__((ext_vector_type(8)));
typedef int          i32x4 __attribute__((ext_vector_type(4)));

// Build TDM descriptor groups for a 2-D tile: tile_dim0 = FEAT dwords,
// tile_dim1 = TILE rows, tensor clamped to rows_rem rows (OOB rows read 0 /
// store dropped).  data_size = 4 bytes (code 2).  See CDNA5 ISA.
__device__ __forceinline__ void tdm_make_desc(uint32_t lds_byte_addr,
                                              const void* global_tile_base,
                                              uint32_t rows_rem,
                                              u32x4& g0, i32x8& g1,
                                              i32x4& g2, i32x4& g3)
{
    uint64_t ga = (uint64_t)(uintptr_t)global_tile_base;
    g0.x = 0x1u;                                         // count=1 (valid user D#)
    g0.y = lds_byte_addr;                                // lds_addr (bytes)
    g0.z = (uint32_t)(ga & 0xFFFFFFFFu);                 // global_addr[31:0]
    g0.w = (uint32_t)((ga >> 32) & 0x01FFFFFFu)          // global_addr[56:32]
         | (2u << 30);                                   // type = 2 ("image")

    const uint32_t dim0   = FEAT;                        // dwords per row
    const uint32_t tdim1  = rows_rem;                    // rows available
    g1[0] = (int)(2u << 16);                             // data_size=2 (4B); mask=0; no pad/iter
    g1[1] = (int)((dim0 & 0xFFFFu) << 16);               // atomic_barrier=0 | tensor_dim0[15:0]
    g1[2] = (int)((dim0 >> 16) | ((tdim1 & 0xFFFFu) << 16)); // dim0[31:16] | dim1[15:0]
    g1[3] = (int)((tdim1 >> 16) | (dim0 << 16));         // dim1[31:16] | tile_dim0
    g1[4] = (int)(TILE & 0xFFFFu);                       // tile_dim1 | tile_dim2=0
    g1[5] = (int)dim0;                                   // tensor_dim0_stride[31:0]
    g1[6] = 0;                                           // stride0[47:32] | stride1[15:0]
    g1[7] = 0;                                           // stride1[47:16]
    g2 = (i32x4){0, 0, 0, 0};                            // dims 2/3 unused (tile_dim3=0)
    g3 = (i32x4){0, 0, 0, 0};                            // dims 4 unused  (tile_dim4=0)
}

__global__ __launch_bounds__(TILE)
void eqnorm_tdm_kernel(const float* __restrict__ x,
                       const float* __restrict__ weight,
                       const float* __restrict__ bias,
                       float* __restrict__ out,
                       int num_atoms)
{
    __shared__ float tile[TILE * FEAT];     // 61440 B
    __shared__ float wb[2 * N_SCALAR];

    const int tid  = threadIdx.x;
    const int base = blockIdx.x * TILE;
    const uint32_t rows_rem = (uint32_t)(num_atoms - base);

    // stage weight/bias (tiny, cached in LDS)
    if (tid < 2 * N_SCALAR)
        wb[tid] = (tid < N_SCALAR) ? weight[tid] : bias[tid - N_SCALAR];

    const uint32_t lds_base = (uint32_t)(uintptr_t)&tile[0];  // flat->LDS offset (addr[31:0])

    // ---- wave 0: async DMA tile of rows into LDS via Tensor Data Mover ----
    if (tid < 32) {
        u32x4 g0; i32x8 g1; i32x4 g2, g3;
        tdm_make_desc(lds_base, x + (size_t)base * FEAT, rows_rem, g0, g1, g2, g3);
        i32x8 gx = (i32x8){0, 0, 0, 0, 0, 0, 0, 0};      // extended group (unused: zeros)
        __builtin_amdgcn_tensor_load_to_lds(g0, g1, g2, g3, gx, 0);
        __builtin_amdgcn_s_wait_tensorcnt(0);
    }
    __syncthreads();

    // ---- one thread per row: normalize in place in LDS ----
    if (base + tid < num_atoms) {
        float* r = &tile[tid * FEAT];

        // LayerNorm over first 32 features
        float s[N_SCALAR];
        float sum = 0.0f;
        #pragma unroll
        for (int k = 0; k < N_SCALAR; ++k) { s[k] = r[k]; sum += s[k]; }
        const float mu = sum * (1.0f / N_SCALAR);
        float var = 0.0f;
        #pragma unroll
        for (int k = 0; k < N_SCALAR; ++k) { float d = s[k] - mu; var += d * d; }
        var *= (1.0f / N_SCALAR);
        const float inv = 1.0f / sqrtf(var + EPS);
        #pragma unroll
        for (int k = 0; k < N_SCALAR; ++k)
            r[k] = (s[k] - mu) * inv * wb[k] + wb[N_SCALAR + k];

        // RMS-norm 16 vectors of dim 3
        #pragma unroll
        for (int j = 0; j < N_VEC; ++j) {
            const int o = N_SCALAR + VEC_DIM * j;
            float a = r[o], bv = r[o + 1], c = r[o + 2];
            float ms = (a * a + bv * bv + c * c) * (1.0f / VEC_DIM);
            float rr = 1.0f / sqrtf(ms + EPS);
            r[o] = a * rr; r[o + 1] = bv * rr; r[o + 2] = c * rr;
        }

        // RMS-norm 8 tensors of dim 5
        #pragma unroll
        for (int j = 0; j < N_TEN; ++j) {
            const int o = N_SCALAR + N_VEC * VEC_DIM + TEN_DIM * j;
            float t0 = r[o], t1 = r[o + 1], t2 = r[o + 2], t3 = r[o + 3], t4 = r[o + 4];
            float ms = (t0 * t0 + t1 * t1 + t2 * t2 + t3 * t3 + t4 * t4) * (1.0f / TEN_DIM);
            float rr = 1.0f / sqrtf(ms + EPS);
            r[o] = t0 * rr; r[o + 1] = t1 * rr; r[o + 2] = t2 * rr;
            r[o + 3] = t3 * rr; r[o + 4] = t4 * rr;
        }
    }
    __syncthreads();

    // ---- wave 0: async DMA tile back out (OOB tail rows dropped by TDM) ----
    if (tid < 32) {
        u32x4 g0; i32x8 g1; i32x4 g2, g3;
        tdm_make_desc(lds_base, out + (size_t)base * FEAT, rows_rem, g0, g1, g2, g3);
        i32x8 gx = (i32x8){0, 0, 0, 0, 0, 0, 0, 0};      // extended group (unused: zeros)
        __builtin_amdgcn_tensor_store_from_lds(g0, g1, g2, g3, gx, 0);
        // S_ENDPGM performs an implicit wait-idle, draining TENSORcnt; the
        // workgroup (and its LDS) stays alive until this wave retires.
    }
}

extern "C" void kernel_launch(void* const* d_in, const int* in_sizes, int n_in,
                              void* d_out, int out_size, void* d_ws, size_t ws_size,
                              hipStream_t stream) {
    const float* x      = (const float*)d_in[0];
    const float* weight = (const float*)d_in[1];
    const float* bias   = (const float*)d_in[2];
    float* out          = (float*)d_out;

    const int num_atoms = in_sizes[0] / FEAT;
    const int grid = (num_atoms + TILE - 1) / TILE;

    hipLaunchKernelGGL(eqnorm_tdm_kernel, dim3(grid), dim3(TILE), 0, stream,
                       x, weight, bias, out, num_atoms);
}